// CrossAttention_60696477827210
// MI455X (gfx1250) — compile-verified
//
#include <hip/hip_runtime.h>

typedef __attribute__((ext_vector_type(2))) float v2f;
typedef __attribute__((ext_vector_type(8))) float v8f;

#define WMMA4F32(ACC, A, B) \
  (ACC) = __builtin_amdgcn_wmma_f32_16x16x4_f32(false, (A), false, (B), (short)0, (ACC), false, false)

// ---------------------------------------------------------------------------
// Generic strided WMMA GEMM:  C[m][n] = alpha * sum_k A[m][k]*B[k][n] + bias[n]
// A is K-contiguous (stride sAm between rows). B element at  k*sBk + n*sBn.
// Batched over grid.z: z -> (zb = z/zdiv, zh = z%zdiv) with per-axis offsets.
// Block = 256 threads = 8 waves; wave (wid) owns one 16x16 tile:
//   block tile = 64 rows x 32 cols, wave tile row = wid&3, col = wid>>2.
// ---------------------------------------------------------------------------
__global__ __launch_bounds__(256) void gemm_wmma_f32(
    const float* __restrict__ A, long sAm,
    const float* __restrict__ B, long sBk, long sBn,
    float* __restrict__ C, long ldC,
    int K, float alpha, const float* __restrict__ bias,
    int zdiv, long sAb, long sAh, long sBb, long sBh, long sCb, long sCh)
{
    const int z  = blockIdx.z;
    const int zb = z / zdiv, zh = z % zdiv;
    A += (long)zb * sAb + (long)zh * sAh;
    B += (long)zb * sBb + (long)zh * sBh;
    C += (long)zb * sCb + (long)zh * sCh;

    const int tid  = threadIdx.x;
    const int lane = tid & 31;
    const int wid  = tid >> 5;
    const int half = lane >> 4;     // 0: K={0,1}, 1: K={2,3} (f32 WMMA A/B layout)
    const int lm   = lane & 15;     // row (A) / col (B,C)

    const int m0 = blockIdx.y * 64 + (wid & 3) * 16;
    const int n0 = blockIdx.x * 32 + (wid >> 2) * 16;

    const float* Ap = A + (long)(m0 + lm) * sAm + 2 * half;
    const float* Bp = B + (long)(n0 + lm) * sBn + (long)(2 * half) * sBk;

    v8f acc = {};
#pragma unroll 4
    for (int kc = 0; kc < K; kc += 4) {
        v2f a = *(const v2f*)Ap;          // A[m][k0], A[m][k0+1] (k-contiguous)
        v2f b;
        b.x = Bp[0];                      // B[k0][n]
        b.y = Bp[sBk];                    // B[k0+1][n]
        WMMA4F32(acc, a, b);
        Ap += 4;
        Bp += 4 * sBk;
    }

    const float bv = bias ? bias[n0 + lm] : 0.0f;
    float* Cp = C + (long)(m0 + 8 * half) * ldC + (n0 + lm);
#pragma unroll
    for (int r = 0; r < 8; ++r)
        Cp[(long)r * ldC] = alpha * acc[r] + bv;   // row m0 + r + 8*half, col n0+lm
}

// ---------------------------------------------------------------------------
// Fused QK^T * scale -> softmax -> attn, one (b, h, 16-row q block) per block.
// S stripe 16 x 1024 lives in LDS (pitch 1028 floats: half-wave row groups
// differ by 8 rows * 4 floats = 32 banks -> conflict-free tile stores).
// q fragments preloaded once per wave (16 x v2f) and reused for all 64 tiles.
// ---------------------------------------------------------------------------
#define SPITCH 1028
#define NEG_INF (-3.402823466e38f)

__global__ __launch_bounds__(256) void attn_softmax_wmma(
    const float* __restrict__ q,    // (B*N1, 1024)   q[b*1024+m][h*64+k]
    const float* __restrict__ kv,   // (B*N2, 2048)   k at [.][h*64+k], v at [.][1024+h*64+d]
    float* __restrict__ attn)       // (B, H, N1, N2)
{
    extern __shared__ float S[];    // 16 * SPITCH floats
    const int b  = blockIdx.z;
    const int h  = blockIdx.y;
    const int m0 = blockIdx.x * 16;

    const int tid  = threadIdx.x;
    const int lane = tid & 31;
    const int wid  = tid >> 5;
    const int half = lane >> 4;
    const int lm   = lane & 15;

    // Preload A (q rows m0..m0+15, hd=64 -> 16 K-steps of 4)
    const float* qbase = q + ((long)(b * 1024 + m0 + lm)) * 1024 + h * 64 + 2 * half;
    v2f afrag[16];
#pragma unroll
    for (int i = 0; i < 16; ++i)
        afrag[i] = *(const v2f*)(qbase + i * 4);

    // B[k][n] = Kmat[n][k]: k-contiguous per lane -> float2 loads
    const float* kbase = kv + (long)b * 1024 * 2048 + h * 64 + 2 * half;

    for (int nt = wid; nt < 64; nt += 8) {
        const int n = nt * 16 + lm;
        const float* Bp = kbase + (long)n * 2048;
        v8f acc = {};
#pragma unroll
        for (int i = 0; i < 16; ++i) {
            v2f bf = *(const v2f*)(Bp + i * 4);
            WMMA4F32(acc, afrag[i], bf);
        }
#pragma unroll
        for (int r = 0; r < 8; ++r)
            S[(r + 8 * half) * SPITCH + n] = acc[r] * 0.125f;   // SCALE = hd^-0.5
    }
    __syncthreads();

    // Softmax: wave wid handles rows wid and wid+8 (32 lanes x 32 elems each)
#pragma unroll
    for (int rr = 0; rr < 2; ++rr) {
        const int row = wid + rr * 8;
        float* Srow = S + row * SPITCH;

        float mx = NEG_INF;
        for (int j = lane; j < 1024; j += 32) mx = fmaxf(mx, Srow[j]);
#pragma unroll
        for (int off = 16; off; off >>= 1) mx = fmaxf(mx, __shfl_xor(mx, off, 32));

        float sum = 0.0f;
        for (int j = lane; j < 1024; j += 32) {
            float e = __expf(Srow[j] - mx);
            Srow[j] = e;
            sum += e;
        }
#pragma unroll
        for (int off = 16; off; off >>= 1) sum += __shfl_xor(sum, off, 32);
        const float inv = 1.0f / sum;

        float* arow = attn + (((long)(b * 16 + h) * 1024) + m0 + row) * 1024;
        for (int j = lane; j < 1024; j += 32) arow[j] = Srow[j] * inv;
    }
}

// ---------------------------------------------------------------------------
extern "C" void kernel_launch(void* const* d_in, const int* in_sizes, int n_in,
                              void* d_out, int out_size, void* d_ws, size_t ws_size,
                              hipStream_t stream)
{
    (void)in_sizes; (void)n_in; (void)out_size; (void)ws_size;

    const float* x     = (const float*)d_in[0];   // (4,1024,1024)
    const float* y     = (const float*)d_in[1];   // (4,1024,1024)
    const float* Wq    = (const float*)d_in[2];   // (1024,1024)
    const float* Wkv   = (const float*)d_in[3];   // (1024,2048)
    const float* Wproj = (const float*)d_in[4];   // (1024,1024)
    const float* bproj = (const float*)d_in[5];   // (1024,)

    float* out  = (float*)d_out;                  // 4*1024*1024
    float* yout = out  + (long)4 * 1024 * 1024;   // 4*1024*1024
    float* attn = yout + (long)4 * 1024 * 1024;   // 4*16*1024*1024

    float* qws  = (float*)d_ws;                   // 4096*1024
    float* kvws = qws  + (long)4096 * 1024;       // 4096*2048
    float* ctx  = kvws + (long)4096 * 2048;       // 4096*1024

    dim3 blk(256, 1, 1);

    // q = x @ Wq                           M=4096 N=1024 K=1024
    gemm_wmma_f32<<<dim3(1024 / 32, 4096 / 64, 1), blk, 0, stream>>>(
        x, 1024, Wq, 1024, 1, qws, 1024, 1024, 1.0f, nullptr,
        1, 0, 0, 0, 0, 0, 0);

    // kv = y @ Wkv                         M=4096 N=2048 K=1024
    gemm_wmma_f32<<<dim3(2048 / 32, 4096 / 64, 1), blk, 0, stream>>>(
        y, 1024, Wkv, 2048, 1, kvws, 2048, 1024, 1.0f, nullptr,
        1, 0, 0, 0, 0, 0, 0);

    // y passthrough (output #2)
    hipMemcpyAsync(yout, y, sizeof(float) * (size_t)4 * 1024 * 1024,
                   hipMemcpyDeviceToDevice, stream);

    // attn = softmax(q k^T * scale)  (output #3)
    attn_softmax_wmma<<<dim3(64, 16, 4), blk, 16 * SPITCH * sizeof(float), stream>>>(
        qws, kvws, attn);

    // ctx = attn @ v, batched z=(b*16+h)   M=1024 N=64 K=1024
    gemm_wmma_f32<<<dim3(64 / 32, 1024 / 64, 64), blk, 0, stream>>>(
        attn, 1024, kvws + 1024, 2048, 1, ctx, 1024, 1024, 1.0f, nullptr,
        16,
        (long)16 * 1024 * 1024, (long)1024 * 1024,   // A: attn (b,h) strides
        (long)1024 * 2048,      64,                  // B: v slice strides
        (long)1024 * 1024,      64);                 // C: ctx (b, h*64) strides

    // out = ctx @ Wproj + bproj  (output #1)   M=4096 N=1024 K=1024
    gemm_wmma_f32<<<dim3(1024 / 32, 4096 / 64, 1), blk, 0, stream>>>(
        ctx, 1024, Wproj, 1024, 1, out, 1024, 1024, 1.0f, bproj,
        1, 0, 0, 0, 0, 0, 0);
}